// CuriosityEngine_42468636623061
// MI455X (gfx1250) — compile-verified
//
#include <hip/hip_runtime.h>
#include <math.h>

// Problem constants (from reference setup_inputs)
#define B_   4
#define T_   2048
#define E_   1536      // 2*n_embd flattened complex features
#define BD_  32        // bond dim
#define H_   256       // surprise hidden
#define NTOK (B_ * T_) // 8192 tokens
#define ROWS_PER_WG 64 // 4 waves x 16-row WMMA strips -> 128 WGs
#define NTHR 128
#define KT   32        // K tile
#define NKT  (E_ / KT) // 48 K tiles
#define XSTR 36        // LDS row stride (144B: 16B-aligned for async B128)

// Workspace layout (float offsets)
#define WS_STATESUM 0     // 4*1536 per-batch column sums of x
#define WS_ENT      6144  // entropy accumulator (identical for all 32 actions)
#define WS_M        6160  // M = Wu @ Wd  (32x32)
#define WS_C        7184  // c = bu @ Wd  (32)

typedef __attribute__((ext_vector_type(2))) float v2f;
typedef __attribute__((ext_vector_type(8))) float v8f;

// ---------------------------------------------------------------------------
// Prep: zero accumulators, precompute M = Wu@Wd (32x32) and c = bu@Wd (32).
// Collapses the 2nd bottleneck pass: z2 = z@M + c  (no 8192x1536 intermediate).
// ---------------------------------------------------------------------------
__global__ __launch_bounds__(256) void prep_kernel(
    const float* __restrict__ Wd, const float* __restrict__ Wu,
    const float* __restrict__ bu, float* __restrict__ ws) {
  const int t = threadIdx.x;
  for (int i = t; i < 6160; i += 256) ws[i] = 0.0f;  // state_sum + ent
  for (int o = t; o < BD_ * BD_; o += 256) {
    const int i = o >> 5, j = o & 31;
    float s = 0.0f;
    for (int e = 0; e < E_; ++e) s = fmaf(Wu[i * E_ + e], Wd[e * BD_ + j], s);
    ws[WS_M + o] = s;
  }
  if (t < BD_) {
    float s = 0.0f;
    for (int e = 0; e < E_; ++e) s = fmaf(bu[e], Wd[e * BD_ + t], s);
    ws[WS_C + t] = s;
  }
}

// ---------------------------------------------------------------------------
// Main fused kernel: one streaming pass over x, double-buffered with
// GLOBAL_LOAD_ASYNC_TO_LDS_B128 (ASYNCcnt) to overlap HBM with WMMA.
//   - z = Xflat @ Wd + bd via v_wmma_f32_16x16x4_f32 (exact f32 matrix math)
//   - fused per-batch column sums of x (for state_mean / surprise head)
//   - per-token entanglement entropy of z2 = z@M + c, reduced to a scalar
// ---------------------------------------------------------------------------
__global__ __launch_bounds__(NTHR) void bottleneck_kernel(
    const float* __restrict__ x, const float* __restrict__ Wd,
    const float* __restrict__ bd, float* __restrict__ ws) {
  __shared__ __align__(16) float Xbuf[2][ROWS_PER_WG * XSTR];  // 2 x 9KB
  __shared__ __align__(16) float Wbuf[2][KT * XSTR];           // 2 x 4.5KB
  __shared__ float Ms[BD_ * BD_];
  __shared__ float cs[BD_];
  __shared__ float bds[BD_];
  __shared__ float red[NTHR];

  const int t = threadIdx.x;
  const int lane = t & 31, w = t >> 5;       // wave 0..3 -> rows w*16..w*16+15
  const int lm = lane & 15, hi = lane >> 4;  // fragment row/col, K-half select
  const int r0 = (int)blockIdx.x * ROWS_PER_WG;
  const int b = r0 / T_;  // 64 | 2048 -> each WG within one batch

  // Per-thread async-copy assignments (byte offsets).
  // X tile: 64 rows x 32 cols = 512 float4; 4 per thread.
  unsigned x_lds[2][4], x_g[4];
#pragma unroll
  for (int i = 0; i < 4; ++i) {
    const int li = t + i * NTHR;          // 0..511
    const int row = li >> 3, c4 = li & 7; // 8 float4 per row
    x_g[i] = (unsigned)(((size_t)(r0 + row) * E_ + c4 * 4) * 4);
    x_lds[0][i] = (unsigned)(size_t)&Xbuf[0][row * XSTR + c4 * 4];
    x_lds[1][i] = (unsigned)(size_t)&Xbuf[1][row * XSTR + c4 * 4];
  }
  // Wd tile: 32 rows x 32 cols = 256 float4 (contiguous 4KB); 2 per thread.
  unsigned w_lds[2][2], w_g[2];
#pragma unroll
  for (int i = 0; i < 2; ++i) {
    const int wj = t + i * NTHR;          // 0..255
    const int row = wj >> 3, c4 = wj & 7;
    w_g[i] = (unsigned)(wj * 16);
    w_lds[0][i] = (unsigned)(size_t)&Wbuf[0][row * XSTR + c4 * 4];
    w_lds[1][i] = (unsigned)(size_t)&Wbuf[1][row * XSTR + c4 * 4];
  }

#define ISSUE_TILE(bufsel, kt2)                                            \
  do {                                                                     \
    const unsigned gx = (unsigned)(kt2) * (KT * 4u);                       \
    const unsigned gw = (unsigned)(kt2) * (KT * BD_ * 4u);                 \
    _Pragma("unroll") for (int i_ = 0; i_ < 4; ++i_)                       \
      asm volatile("global_load_async_to_lds_b128 %0, %1, %2"              \
                   :: "v"(x_lds[bufsel][i_]), "v"(x_g[i_] + gx), "s"(x)    \
                   : "memory");                                            \
    _Pragma("unroll") for (int i_ = 0; i_ < 2; ++i_)                       \
      asm volatile("global_load_async_to_lds_b128 %0, %1, %2"              \
                   :: "v"(w_lds[bufsel][i_]), "v"(w_g[i_] + gw), "s"(Wd)   \
                   : "memory");                                            \
  } while (0)

  // Prologue: kick off tile 0, load small constants meanwhile.
  ISSUE_TILE(0, 0);
  for (int i = t; i < BD_ * BD_; i += NTHR) Ms[i] = ws[WS_M + i];
  if (t < BD_) { cs[t] = ws[WS_C + t]; bds[t] = bd[t]; }

  v8f acc0 = {};  // columns 0..15
  v8f acc1 = {};  // columns 16..31
  float* statesum = ws + WS_STATESUM;

  for (int kt = 0; kt < NKT; ++kt) {
    const int cur = kt & 1;
    // Prefetch next tile into the other buffer (its readers finished last iter)
    if (kt + 1 < NKT) {
      ISSUE_TILE(1 - cur, kt + 1);
      asm volatile("s_wait_asynccnt 0x6" ::: "memory");  // tile kt landed (in-order)
    } else {
      asm volatile("s_wait_asynccnt 0x0" ::: "memory");
    }
    __syncthreads();  // all waves' tile-kt data visible in LDS

    const float* Xs = Xbuf[cur];
    const float* Wb = Wbuf[cur];

    // 8 WMMA K=4 steps; ISA f32 A-frag: lanes 0-15 hold K0/K1, 16-31 hold K2/K3
#pragma unroll
    for (int kk = 0; kk < 8; ++kk) {
      v2f a, b0v, b1v;
      const float* ap = &Xs[(w * 16 + lm) * XSTR + kk * 4 + 2 * hi];
      a.x = ap[0]; a.y = ap[1];
      const int kr = kk * 4 + 2 * hi;
      b0v.x = Wb[kr * XSTR + lm];       b0v.y = Wb[(kr + 1) * XSTR + lm];
      b1v.x = Wb[kr * XSTR + 16 + lm];  b1v.y = Wb[(kr + 1) * XSTR + 16 + lm];
      acc0 = __builtin_amdgcn_wmma_f32_16x16x4_f32(false, a, false, b0v,
                                                   (short)0, acc0, false, false);
      acc1 = __builtin_amdgcn_wmma_f32_16x16x4_f32(false, a, false, b1v,
                                                   (short)0, acc1, false, false);
    }

    // Fused per-batch column sums (state_mean numerator) over the tile in LDS
    {
      const int col = t & 31, seg = t >> 5;  // 4 segments of 16 rows
      float s = 0.0f;
#pragma unroll
      for (int r = 0; r < 16; ++r) s += Xs[(seg * 16 + r) * XSTR + col];
      red[t] = s;
    }
    __syncthreads();
    if (t < 32) {
      float s = red[t] + red[32 + t] + red[64 + t] + red[96 + t];
      atomicAdd(&statesum[b * E_ + kt * KT + t], s);
    }
    __syncthreads();  // readers of Xs/Wb/red done -> buffer reusable
  }

  // Spill z tile (+bd) to LDS so each thread sees a full 32-wide token row.
  // D-matrix layout: VGPR r <-> rows r (lanes 0-15) and r+8 (lanes 16-31).
  float* zs = Xbuf[0];
#pragma unroll
  for (int r = 0; r < 8; ++r) {
    const int row = w * 16 + r + hi * 8;
    zs[row * XSTR + lm]      = acc0[r] + bds[lm];
    zs[row * XSTR + 16 + lm] = acc1[r] + bds[16 + lm];
  }
  __syncthreads();

  // Per-token entropy of p = z2^2 / sum(z2^2), z2 = z @ M + c
  float ent = 0.0f;
  if (t < ROWS_PER_WG) {
    float zr[32], z2v[32];
#pragma unroll
    for (int i = 0; i < 32; ++i) zr[i] = zs[t * XSTR + i];
    float s2 = 0.0f;
    for (int j = 0; j < 32; ++j) {
      float z2 = cs[j];
#pragma unroll
      for (int i = 0; i < 32; ++i) z2 = fmaf(zr[i], Ms[i * 32 + j], z2);
      z2v[j] = z2;
      s2 = fmaf(z2, z2, s2);
    }
    s2 = fmaxf(s2, 1e-12f);
    const float inv = 1.0f / s2;
    for (int j = 0; j < 32; ++j) {
      const float p = z2v[j] * z2v[j] * inv;
      ent -= p * __logf(p + 1e-12f);
    }
  }
  red[t] = (t < ROWS_PER_WG) ? ent : 0.0f;
  __syncthreads();
  if (t == 0) {
    float s = 0.0f;
    for (int i = 0; i < ROWS_PER_WG; ++i) s += red[i];
    atomicAdd(&ws[WS_ENT], s);  // identical for every action replica
  }
#undef ISSUE_TILE
}

// ---------------------------------------------------------------------------
// Surprise head: state_mean -> Linear(1536,256) -> ReLU -> Linear(256,1).
// best_action_idx: all 32 action replicas are bitwise identical (pure
// broadcast), so argmax of a constant vector == 0.
// ---------------------------------------------------------------------------
__global__ __launch_bounds__(256) void surprise_kernel(
    const float* __restrict__ W1, const float* __restrict__ b1,
    const float* __restrict__ W2, const float* __restrict__ b2,
    const float* __restrict__ ws, float* __restrict__ out) {
  __shared__ float hs[B_ * H_];
  const int t = threadIdx.x;  // t == hidden index j
  const float invT = 1.0f / (float)T_;
  for (int b = 0; b < B_; ++b) {
    float h = b1[t];
    for (int e = 0; e < E_; ++e)
      h = fmaf(ws[WS_STATESUM + b * E_ + e] * invT, W1[e * H_ + t], h);
    hs[b * H_ + t] = fmaxf(h, 0.0f);
  }
  __syncthreads();
  if (t < B_) {
    float s = b2[0];
    for (int j = 0; j < H_; ++j) s = fmaf(hs[t * H_ + j], W2[j], s);
    out[t] = s;
  }
  if (t == B_) out[B_] = 0.0f;  // best_action_idx
}

extern "C" void kernel_launch(void* const* d_in, const int* in_sizes, int n_in,
                              void* d_out, int out_size, void* d_ws, size_t ws_size,
                              hipStream_t stream) {
  (void)in_sizes; (void)n_in; (void)out_size; (void)ws_size;
  const float* x  = (const float*)d_in[0];
  const float* W1 = (const float*)d_in[1];
  const float* b1 = (const float*)d_in[2];
  const float* W2 = (const float*)d_in[3];
  const float* b2 = (const float*)d_in[4];
  const float* Wd = (const float*)d_in[5];
  const float* bd = (const float*)d_in[6];
  const float* Wu = (const float*)d_in[7];
  const float* bu = (const float*)d_in[8];
  float* ws  = (float*)d_ws;
  float* out = (float*)d_out;

  hipLaunchKernelGGL(prep_kernel, dim3(1), dim3(256), 0, stream, Wd, Wu, bu, ws);
  hipLaunchKernelGGL(bottleneck_kernel, dim3(NTOK / ROWS_PER_WG), dim3(NTHR), 0, stream,
                     x, Wd, bd, ws);
  hipLaunchKernelGGL(surprise_kernel, dim3(1), dim3(256), 0, stream,
                     W1, b1, W2, b2, ws, out);
}